// AuxiliaryDenseCriterion_72653666779385
// MI455X (gfx1250) — compile-verified
//
#include <hip/hip_runtime.h>
#include <hip/hip_bf16.h>
#include <cstdint>

#define BATCH 16
#define NLOC  21504
#define NCLS  80
#define GDIM  64
#define KTOP  9
#define ALPHA_F 0.25f

#define NCHUNK 21
#define CHUNK  (NLOC / NCHUNK)          // 1024
#define TILES_PER_CHUNK (CHUNK / 16)    // 64 (power of two)

typedef float v2f __attribute__((ext_vector_type(2)));
typedef float v8f __attribute__((ext_vector_type(8)));

// Sorted ascending top-9 insert on u64 keys (key = dsq_bits<<32 | idx, unique).
__device__ __forceinline__ void top9_insert(unsigned long long* top,
                                            unsigned long long key)
{
  if (key < top[KTOP - 1]) {
    bool bet[KTOP];  // entry k better (smaller) than candidate
#pragma unroll
    for (int k = 0; k < KTOP; ++k) bet[k] = top[k] < key;
#pragma unroll
    for (int k = KTOP - 1; k > 0; --k)
      if (!bet[k - 1]) top[k] = top[k - 1];
#pragma unroll
    for (int k = 0; k < KTOP; ++k)
      if (!bet[k] && (k == 0 || bet[k - 1])) top[k] = key;
  }
}

// ---------------------------------------------------------------------------
// Kernel 1: squared-distance tiles via V_WMMA_F32_16X16X4_F32, top-9 fully in
// registers.  D = A x B with
//   A row (location m) = [lx, ly, 1, lx^2+ly^2]        (per-tile, 16 locs)
//   B col (GT n)       = [-2cx, -2cy, cx^2+cy^2, 1]    (loop-invariant)
//   D[m][n] = (cx-lx)^2 + (cy-ly)^2
// D layout: lane n holds column gt=n, vgpr v = loc v (lanes 16-31: loc v+8)
// => each lane scans its 8 accumulator registers for its own GT; no LDS in loop.
// Location loads are software-pipelined one tile ahead (+ prefetch 2 ahead) so
// the WMMA + scan of tile t overlaps the fetch of tile t+1.
// Grid: (NCHUNK, 64 GT-strips), 1 wave per block.
// ---------------------------------------------------------------------------
__global__ __launch_bounds__(32) void adc_dist_topk_kernel(
    const float* __restrict__ gt_boxes,     // [B,G,4] cxcywh
    const float* __restrict__ locations,    // [N,2]
    unsigned long long* __restrict__ cand)  // [B*G][NCHUNK][KTOP]
{
  __shared__ unsigned long long mbuf[16 * KTOP];
  const int lane  = threadIdx.x;
  const int half  = lane >> 4;
  const int l16   = lane & 15;
  const int chunk = blockIdx.x;           // 0..NCHUNK-1
  const int row0  = blockIdx.y * 16;      // GT strip base (B*G = 1024 rows)
  const int base  = chunk * CHUNK;

  // Loop-invariant B operand: this lane's GT column.
  const int   r   = row0 + l16;
  const float cx  = gt_boxes[r * 4 + 0];
  const float cy  = gt_boxes[r * 4 + 1];
  const float csq = cx * cx + cy * cy;
  v2f bm;
  bm[0] = half ? csq  : (-2.0f * cx);
  bm[1] = half ? 1.0f : (-2.0f * cy);

  unsigned long long top[KTOP];
#pragma unroll
  for (int k = 0; k < KTOP; ++k) top[k] = ~0ULL;

  // Pipeline prologue: fetch tile 0's location for this lane.
  float lx = locations[(base + l16) * 2 + 0];
  float ly = locations[(base + l16) * 2 + 1];

  for (int t = 0; t < TILES_PER_CHUNK; ++t) {
    // Issue next tile's load before consuming the current one (wrap keeps the
    // address in range without a branch; last iteration's extra load is dead).
    const int jn = base + ((t + 1) & (TILES_PER_CHUNK - 1)) * 16 + l16;
    const float nlx = locations[jn * 2 + 0];
    const float nly = locations[jn * 2 + 1];
    __builtin_prefetch(&locations[(base + ((t + 2) & (TILES_PER_CHUNK - 1)) * 16 + l16) * 2], 0, 3);

    v2f a;
    a[0] = half ? 1.0f : lx;
    a[1] = half ? (lx * lx + ly * ly) : ly;

    v8f c = {};
    c = __builtin_amdgcn_wmma_f32_16x16x4_f32(
        /*neg_a=*/false, a, /*neg_b=*/false, bm,
        /*c_mod=*/(short)0, c, /*reuse_a=*/false, /*reuse_b=*/false);

#pragma unroll
    for (int v = 0; v < 8; ++v) {
      const float    dsq = fmaxf(c[v], 0.0f);   // clamp rounding negatives
      const unsigned idx = (unsigned)(base + t * 16 + half * 8 + v);
      const unsigned long long key =
          ((unsigned long long)__float_as_uint(dsq) << 32) | idx;
      top9_insert(top, key);
    }
    lx = nlx;
    ly = nly;
  }

  // Merge the two half-wave lists for each GT, write chunk candidates.
  if (half == 1) {
#pragma unroll
    for (int k = 0; k < KTOP; ++k) mbuf[l16 * KTOP + k] = top[k];
  }
  __syncthreads();
  if (half == 0) {
#pragma unroll
    for (int k = 0; k < KTOP; ++k) top9_insert(top, mbuf[l16 * KTOP + k]);
    unsigned long long* dst =
        &cand[((unsigned)(row0 + l16) * NCHUNK + chunk) * KTOP];
#pragma unroll
    for (int k = 0; k < KTOP; ++k) dst[k] = top[k];
  }
}

// ---------------------------------------------------------------------------
// Kernel 2: merge per-chunk top-9 lists -> final top-9 indices per GT row.
// ---------------------------------------------------------------------------
__global__ __launch_bounds__(256) void adc_merge_kernel(
    const unsigned long long* __restrict__ cand, int* __restrict__ topk_idx)
{
  const int r = blockIdx.x * blockDim.x + threadIdx.x;
  if (r >= BATCH * GDIM) return;
  unsigned long long top[KTOP];
#pragma unroll
  for (int k = 0; k < KTOP; ++k) top[k] = ~0ULL;
  const unsigned long long* src = &cand[(unsigned)r * NCHUNK * KTOP];
  for (int i = 0; i < NCHUNK * KTOP; ++i) top9_insert(top, src[i]);
#pragma unroll
  for (int k = 0; k < KTOP; ++k)
    topk_idx[r * KTOP + k] = (int)(unsigned)(top[k] & 0xffffffffULL);
}

// ---------------------------------------------------------------------------
// Kernel 3: zero the per-(b,n) class bitmask (3 uint32 words per row).
// ---------------------------------------------------------------------------
__global__ __launch_bounds__(256) void adc_zero_kernel(uint32_t* __restrict__ p, int n)
{
  const int i = blockIdx.x * blockDim.x + threadIdx.x;
  if (i < n) p[i] = 0u;
}

// ---------------------------------------------------------------------------
// Kernel 4: scatter sparse targets (atomicOr dedups like .at[].set(1.0),
// order-independent -> deterministic).
// ---------------------------------------------------------------------------
__global__ __launch_bounds__(256) void adc_scatter_kernel(
    const int* __restrict__ topk_idx, const int* __restrict__ gt_labels,
    uint32_t* __restrict__ mask)
{
  const int e = blockIdx.x * blockDim.x + threadIdx.x;  // [0, B*G*KTOP)
  if (e >= BATCH * GDIM * KTOP) return;
  const int rg = e / KTOP;
  const int b  = rg / GDIM;
  const int n  = topk_idx[e];
  const int c  = gt_labels[rg];
  atomicOr(&mask[(b * NLOC + n) * 3 + (c >> 5)], 1u << (c & 31));
}

// ---------------------------------------------------------------------------
// Deterministic block tree-reduce.
// ---------------------------------------------------------------------------
__device__ __forceinline__ float block_reduce_sum(float v, float* red)
{
  const int t = threadIdx.x;
  red[t] = v;
  __syncthreads();
  for (int s = blockDim.x >> 1; s > 0; s >>= 1) {
    if (t < s) red[t] += red[t + s];
    __syncthreads();
  }
  const float r = red[0];
  __syncthreads();
  return r;
}

// ---------------------------------------------------------------------------
// Kernel 5: fused sigmoid focal loss streaming the 110 MB logits once.
// Hardware transcendentals (v_exp_f32 / v_log_f32 / v_rcp_f32) keep this near
// the HBM roofline instead of 4x over it with libm expf/log1pf:
//   z  = e^{-|x|}                    (one v_exp)
//   ce = softplus(+-x) = max(ax,0) + log(1+z)     (one v_log)
//   pr = sigmoid(x) from rcp(1+z)    (one v_rcp)
// log(1+z) vs log1p(z) only differs for z < 2^-24 where the focal term is
// negligible in the 27.5M-element mean.
// ---------------------------------------------------------------------------
__global__ __launch_bounds__(256) void adc_focal_kernel(
    const float* __restrict__ logits, const uint32_t* __restrict__ mask,
    float* __restrict__ partial)
{
  __shared__ float red[256];
  const int R = blockIdx.x * 32;
  float s = 0.0f;
  for (int i = threadIdx.x; i < 32 * NCLS; i += 256) {
    const int row = R + i / NCLS;
    const int c   = i - (i / NCLS) * NCLS;
    const float    x  = logits[row * NCLS + c];
    const uint32_t mw = mask[row * 3 + (c >> 5)];
    const bool t1 = (mw >> (c & 31)) & 1u;

    const float z    = __expf(-fabsf(x));
    const float opz  = 1.0f + z;
    const float l1z  = __logf(opz);
    const float rz   = __builtin_amdgcn_rcpf(opz);
    const float ax   = t1 ? -x : x;
    const float ce   = fmaxf(ax, 0.0f) + l1z;                 // softplus(ax)
    const float pr   = (x >= 0.0f) ? rz : z * rz;             // sigmoid(x)
    const float ompt = t1 ? (1.0f - pr) : pr;                 // 1 - p_t
    const float w    = t1 ? ALPHA_F : (1.0f - ALPHA_F);
    s += w * ce * ompt * ompt;
  }
  const float bs = block_reduce_sum(s, red);
  if (threadIdx.x == 0) partial[blockIdx.x] = bs;
}

// ---------------------------------------------------------------------------
// Kernel 6: L1 + GIoU over the 9216 gathered (pred, gt) pairs.
// ---------------------------------------------------------------------------
__global__ __launch_bounds__(256) void adc_bbox_kernel(
    const float* __restrict__ pred_boxes, const float* __restrict__ gt_boxes,
    const int* __restrict__ topk_idx, float* __restrict__ partial2)
{
  __shared__ float red[256];
  const int e = blockIdx.x * blockDim.x + threadIdx.x;
  float sab = 0.0f, sg = 0.0f;
  if (e < BATCH * GDIM * KTOP) {
    const int rg = e / KTOP;
    const int b  = rg / GDIM;
    const int n  = topk_idx[e];
    const float* pb = &pred_boxes[(b * NLOC + n) * 4];
    const float* gb = &gt_boxes[rg * 4];
    const float pcx = pb[0], pcy = pb[1], pw = pb[2], ph = pb[3];
    const float gcx = gb[0], gcy = gb[1], gw = gb[2], gh = gb[3];
    sab = fabsf(pcx - gcx) + fabsf(pcy - gcy) + fabsf(pw - gw) + fabsf(ph - gh);
    const float ax0 = pcx - 0.5f * pw, ay0 = pcy - 0.5f * ph;
    const float ax1 = pcx + 0.5f * pw, ay1 = pcy + 0.5f * ph;
    const float bx0 = gcx - 0.5f * gw, by0 = gcy - 0.5f * gh;
    const float bx1 = gcx + 0.5f * gw, by1 = gcy + 0.5f * gh;
    const float area_a = (ax1 - ax0) * (ay1 - ay0);
    const float area_b = (bx1 - bx0) * (by1 - by0);
    const float iw = fmaxf(fminf(ax1, bx1) - fmaxf(ax0, bx0), 0.0f);
    const float ih = fmaxf(fminf(ay1, by1) - fmaxf(ay0, by0), 0.0f);
    const float inter = iw * ih;
    const float uni   = area_a + area_b - inter;
    const float iou   = inter / uni;
    const float cw = fmaxf(fmaxf(ax1, bx1) - fminf(ax0, bx0), 0.0f);
    const float ch = fmaxf(fmaxf(ay1, by1) - fminf(ay0, by0), 0.0f);
    const float area_c = cw * ch;
    const float giou = iou - (area_c - uni) / area_c;
    sg = 1.0f - giou;
  }
  const float s0 = block_reduce_sum(sab, red);
  const float s1 = block_reduce_sum(sg, red);
  if (threadIdx.x == 0) {
    partial2[blockIdx.x * 2 + 0] = s0;
    partial2[blockIdx.x * 2 + 1] = s1;
  }
}

// ---------------------------------------------------------------------------
// Kernel 7: final deterministic reduction -> 3 output scalars.
// ---------------------------------------------------------------------------
__global__ __launch_bounds__(256) void adc_final_kernel(
    const float* __restrict__ fpart, int nf,
    const float* __restrict__ bpart, int nb,
    float* __restrict__ out)
{
  __shared__ float red[256];
  float s = 0.0f;
  for (int i = threadIdx.x; i < nf; i += 256) s += fpart[i];
  const float cls_sum = block_reduce_sum(s, red);
  s = 0.0f;
  for (int i = threadIdx.x; i < nb; i += 256) s += bpart[2 * i + 0];
  const float ab_sum = block_reduce_sum(s, red);
  s = 0.0f;
  for (int i = threadIdx.x; i < nb; i += 256) s += bpart[2 * i + 1];
  const float g_sum = block_reduce_sum(s, red);
  if (threadIdx.x == 0) {
    out[0] = cls_sum / (float)(BATCH * NLOC * NCLS);
    out[1] = ab_sum  / (float)(BATCH * GDIM * KTOP * 4);
    out[2] = g_sum   / (float)(BATCH * GDIM * KTOP);
  }
}

// ---------------------------------------------------------------------------
extern "C" void kernel_launch(void* const* d_in, const int* in_sizes, int n_in,
                              void* d_out, int out_size, void* d_ws, size_t ws_size,
                              hipStream_t stream)
{
  (void)in_sizes; (void)n_in; (void)out_size; (void)ws_size;
  const float* pred_logits = (const float*)d_in[0];   // [B,N,C]
  const float* pred_boxes  = (const float*)d_in[1];   // [B,N,4]
  const float* locations   = (const float*)d_in[2];   // [N,2]
  const float* gt_boxes    = (const float*)d_in[3];   // [B,G,4]
  const int*   gt_labels   = (const int*)d_in[4];     // [B,G]
  float* out = (float*)d_out;

  // Workspace layout (256B-aligned sections, ~5.76 MB total):
  char* ws = (char*)d_ws;
  unsigned long long* cand = (unsigned long long*)(ws);          // 1,548,288 B
  int*      topk  = (int*)(ws + 1548288);                        //    36,864 B
  uint32_t* mask  = (uint32_t*)(ws + 1548288 + 36864);           // 4,128,768 B
  float*    fpart = (float*)(ws + 1548288 + 36864 + 4128768);    //    43,008 B
  float*    bpart = (float*)(ws + 1548288 + 36864 + 4128768 + 43008);  // 288 B

  const int n_mask    = BATCH * NLOC * 3;            // 1,032,192 words
  const int n_entries = BATCH * GDIM * KTOP;         // 9,216
  const int nb_bbox   = (n_entries + 255) / 256;     // 36
  const int nb_focal  = (BATCH * NLOC) / 32;         // 10,752

  dim3 topk_grid(NCHUNK, (BATCH * GDIM) / 16);       // 21 x 64 waves
  adc_dist_topk_kernel<<<topk_grid, 32, 0, stream>>>(gt_boxes, locations, cand);
  adc_merge_kernel  <<<(BATCH * GDIM + 255) / 256, 256, 0, stream>>>(cand, topk);
  adc_zero_kernel   <<<(n_mask + 255) / 256, 256, 0, stream>>>(mask, n_mask);
  adc_scatter_kernel<<<(n_entries + 255) / 256, 256, 0, stream>>>(topk, gt_labels, mask);
  adc_focal_kernel  <<<nb_focal, 256, 0, stream>>>(pred_logits, mask, fpart);
  adc_bbox_kernel   <<<nb_bbox, 256, 0, stream>>>(pred_boxes, gt_boxes, topk, bpart);
  adc_final_kernel  <<<1, 256, 0, stream>>>(fpart, nb_focal, bpart, nb_bbox, out);
}